// VectorQuantizer_75419625718679
// MI455X (gfx1250) — compile-verified
//
#include <hip/hip_runtime.h>

typedef float v2f __attribute__((ext_vector_type(2)));
typedef float v8f __attribute__((ext_vector_type(8)));

#define NUM_EMB   1024
#define EMB_DIM   64
#define N_VEC     131072            // 32 * 64 * 64
#define Z_ELEMS   8388608           // N_VEC * EMB_DIM

// ---------------------------------------------------------------------------
// Kernel 0: ||e||^2 per codebook row -> ws[64 + e]; also zero the loss
// accumulator ws[0] (harness poisons d_ws, we must init every call).
// ---------------------------------------------------------------------------
__global__ void vq_enorm_kernel(const float* __restrict__ emb,
                                float* __restrict__ ws) {
    int e = blockIdx.x * blockDim.x + threadIdx.x;
    if (e == 0) ws[0] = 0.0f;
    if (e < NUM_EMB) {
        const float4* row = (const float4*)(emb + e * EMB_DIM);
        float s = 0.0f;
#pragma unroll
        for (int i = 0; i < EMB_DIM / 4; ++i) {
            float4 v = row[i];
            s += v.x * v.x + v.y * v.y + v.z * v.z + v.w * v.w;
        }
        ws[64 + e] = s;
    }
}

// ---------------------------------------------------------------------------
// Kernel 1: distance argmin via V_WMMA_F32_16X16X4_F32.
// One wave = 16 rows of z_flat. Loop over 64 tiles of 16 embeddings,
// accumulating z.e^T over K=64 in 16 WMMA steps (fp32 in, fp32 acc).
// score = ||e||^2 - 2*dot (||z||^2 constant per row, irrelevant to argmin).
// ---------------------------------------------------------------------------
__global__ __launch_bounds__(256)
void vq_argmin_kernel(const float* __restrict__ z,
                      const float* __restrict__ emb,
                      const float* __restrict__ enorm,
                      int* __restrict__ idx_out) {
    const int lane  = threadIdx.x & 31;
    const int wave  = threadIdx.x >> 5;
    const int gw    = blockIdx.x * 8 + wave;   // global wave id
    const int n0    = gw * 16;                 // first flat row of this tile

    // n = b*4096 + h*64 + w  (z_flat row);  z[b,c,h,w] channel stride = 4096
    const int b  = n0 >> 12;
    const int h  = (n0 >> 6) & 63;
    const int w0 = n0 & 63;                    // multiple of 16
    const float* zb = z + b * 262144 + h * 64 + w0;  // &z[b,0,h,w0]

    const int m     = lane & 15;   // row within tile (A) / emb within tile (B)
    const int khalf = lane >> 4;   // K-half select per ISA 16x4 A/B layout

    // A tile: 16 k-steps; step k: VGPR0 = K=4k+2*khalf, VGPR1 = +1
    v2f a[16];
#pragma unroll
    for (int k = 0; k < 16; ++k) {
        const int c = 4 * k + 2 * khalf;
        a[k].x = zb[(c + 0) * 4096 + m];
        a[k].y = zb[(c + 1) * 4096 + m];
    }

    float best[8];
    int   bidx[8];
#pragma unroll
    for (int v = 0; v < 8; ++v) { best[v] = 3.0e38f; bidx[v] = 0; }

    for (int t = 0; t < 64; ++t) {
        const int e0 = t * 16;
        const int e  = e0 + m;                       // this lane's candidate
        const float* ept = emb + e * EMB_DIM + 2 * khalf;
        if (t < 63)   // gfx1250 global_prefetch_b8 for the next emb tile
            __builtin_prefetch(ept + 16 * EMB_DIM, 0, 0);

        v8f acc = {0.f, 0.f, 0.f, 0.f, 0.f, 0.f, 0.f, 0.f};
#pragma unroll
        for (int k = 0; k < 16; ++k) {
            v2f bv = *(const v2f*)(ept + 4 * k);     // contiguous {c, c+1}
            acc = __builtin_amdgcn_wmma_f32_16x16x4_f32(
                /*neg_a=*/false, a[k], /*neg_b=*/false, bv,
                /*c_mod=*/(short)0, acc, /*reuse_a=*/false, /*reuse_b=*/false);
        }

        const float en = enorm[e];
#pragma unroll
        for (int v = 0; v < 8; ++v) {
            const float score = en - 2.0f * acc[v];  // row M = v + 8*khalf
            if (score < best[v]) { best[v] = score; bidx[v] = e; }
        }
    }

    // Reduce argmin across the 16 lanes of each half-wave (N dimension),
    // tie-break to the lowest embedding index (matches jnp.argmin).
#pragma unroll
    for (int v = 0; v < 8; ++v) {
#pragma unroll
        for (int off = 1; off < 16; off <<= 1) {
            float ob = __shfl_xor(best[v], off, 16);
            int   oi = __shfl_xor(bidx[v], off, 16);
            if (ob < best[v] || (ob == best[v] && oi < bidx[v])) {
                best[v] = ob; bidx[v] = oi;
            }
        }
    }

    if (m == 0) {
#pragma unroll
        for (int v = 0; v < 8; ++v)
            idx_out[n0 + khalf * 8 + v] = bidx[v];
    }
}

// ---------------------------------------------------------------------------
// Kernel 2: quantized = emb[idx[j/64]][j%64] (faithful flat-reshape, layout
// bug preserved) fused with sum((z - quantized)^2) partial reduction.
// ---------------------------------------------------------------------------
__global__ __launch_bounds__(256)
void vq_quant_loss_kernel(const float* __restrict__ z,
                          const float* __restrict__ emb,
                          const int* __restrict__ idx,
                          float* __restrict__ outq,
                          float* __restrict__ sum_ws) {
    __shared__ float red[256];
    const int t = blockIdx.x * 256 + threadIdx.x;
    const int j = t * 4;                       // float4 per thread
    const int r = j >> 6;                      // flat row (shares one index)
    const int c = j & 63;

    const int e = idx[r];
    float4 q  = *(const float4*)(emb + e * EMB_DIM + c);
    float4 zv = *(const float4*)(z + j);
    *(float4*)(outq + j) = q;                  // straight-through value == q

    const float dx = zv.x - q.x, dy = zv.y - q.y;
    const float dz = zv.z - q.z, dw = zv.w - q.w;
    red[threadIdx.x] = dx * dx + dy * dy + dz * dz + dw * dw;
    __syncthreads();
#pragma unroll
    for (int off = 128; off > 0; off >>= 1) {
        if (threadIdx.x < off) red[threadIdx.x] += red[threadIdx.x + off];
        __syncthreads();
    }
    if (threadIdx.x == 0) atomicAdd(sum_ws, red[0]);
}

// ---------------------------------------------------------------------------
// Kernel 3: loss = (1 + COMMIT) * mean(sq diff)   (both terms equal in value)
// ---------------------------------------------------------------------------
__global__ void vq_finalize_kernel(const float* __restrict__ ws,
                                   float* __restrict__ loss_out) {
    loss_out[0] = 1.25f * ws[0] * (1.0f / (float)Z_ELEMS);
}

// ---------------------------------------------------------------------------
// d_out layout: [0, 8388608) quantized f32 | [8388608] loss f32 |
//               [8388609, 8519681) encoding indices (int32 values)
// d_ws layout:  ws[0] = loss accumulator | ws[64..64+1023] = ||e||^2
// ---------------------------------------------------------------------------
extern "C" void kernel_launch(void* const* d_in, const int* in_sizes, int n_in,
                              void* d_out, int out_size, void* d_ws, size_t ws_size,
                              hipStream_t stream) {
    const float* z   = (const float*)d_in[0];
    const float* emb = (const float*)d_in[1];

    float* ws    = (float*)d_ws;
    float* outq  = (float*)d_out;
    float* lossp = (float*)d_out + Z_ELEMS;
    int*   idxp  = (int*)d_out + (Z_ELEMS + 1);

    vq_enorm_kernel<<<NUM_EMB / 256, 256, 0, stream>>>(emb, ws);
    vq_argmin_kernel<<<N_VEC / (16 * 8), 256, 0, stream>>>(z, emb, ws + 64, idxp);
    vq_quant_loss_kernel<<<Z_ELEMS / (4 * 256), 256, 0, stream>>>(z, emb, idxp, outq, ws);
    vq_finalize_kernel<<<1, 1, 0, stream>>>(ws, lossp);
}